// MemoryTree_53317724013012
// MI455X (gfx1250) — compile-verified
//
#include <hip/hip_runtime.h>
#include <hip/hip_bf16.h>

// ---------------------------------------------------------------------------
// MemoryTree oracle on MI455X (gfx1250, wave32, WMMA)
//
// Key identity: _fusion ignores x_shared, so hm[k][b,n] = 2^-k * sum of the
// 2^k leaf matrices under node n.  The descent comparison sL < sR is scale
// invariant per level, so it equals comparing SUMS of per-leaf quadratic
// forms s[b,leaf,j] = q_j^T M_leaf q_j.
//
// Phase A: s[b,leaf,j] via fp32 WMMA (V_WMMA_F32_16X16X4_F32):
//            T = M(64x64) x Q^T(64x256), s = colwise dot with Q.
// Phase B: hierarchical pair-sums of s over leaves (levels 1..9).
// Phase C: per-(b,j) descent comparing sums -> leaf index.
// ---------------------------------------------------------------------------

typedef float v2f __attribute__((ext_vector_type(2)));
typedef float v8f __attribute__((ext_vector_type(8)));

#define B_DIM  8
#define L_DIM  1024
#define D_DIM  64
#define LK_DIM 256

// workspace layout (floats):
//   [0, S0_FLOATS)                : leaf scores S0[b][leaf][j]
//   [S0_FLOATS, +TREE_FLOATS)     : sum-tree levels 1..9, level k = [b][n][j]
#define S0_FLOATS  ((size_t)B_DIM * L_DIM * LK_DIM)   // 2,097,152
#define TREE_FLOATS ((size_t)2093056)

__device__ __forceinline__ size_t lvl_off(int k) {
  // cumulative start offsets (floats) of level k (k=1..9) inside tree region
  const size_t offs[10] = {0, 0, 1048576, 1572864, 1835008,
                           1966080, 2031616, 2064384, 2080768, 2088960};
  return offs[k];
}

// ---------------------------------------------------------------------------
// Phase A: per-leaf quadratic-form scores.
// grid = 1024 blocks x 256 threads (8 waves).  block -> (b, group of 8 leaves)
// LDS: full Q[b] (256x64 f32 = 64 KB), staged once per block.
// ---------------------------------------------------------------------------
__global__ void __launch_bounds__(256)
mt_score_kernel(const float* __restrict__ leafs,
                const float* __restrict__ q,
                float* __restrict__ S0)
{
  __shared__ __align__(16) float Qs[LK_DIM * D_DIM];   // 64 KB

  const int b        = blockIdx.x >> 7;          // 128 blocks per batch
  const int leafBase = (blockIdx.x & 127) * 8;   // 8 leaves per block

  // ---- stage Q[b] into LDS (256 threads x 16 float4) ----
  const float* Qg = q + (size_t)b * LK_DIM * D_DIM;
  for (int i = threadIdx.x; i < (LK_DIM * D_DIM) / 4; i += blockDim.x) {
    ((float4*)Qs)[i] = ((const float4*)Qg)[i];
  }
  __syncthreads();

  const int wave = threadIdx.x >> 5;   // wave32
  const int lane = threadIdx.x & 31;
  const int lr   = lane & 15;          // row-within-tile / column-within-tile
  const int hi   = lane >> 4;          // lane half: selects K pair / row half
  const int leaf = leafBase + wave;

  const float* __restrict__ M =
      leafs + ((size_t)b * L_DIM + leaf) * D_DIM * D_DIM;

#if __has_builtin(__builtin_amdgcn_wmma_f32_16x16x4_f32)
  // ---- preload the whole 64x64 M as WMMA A-fragments -------------------
  // ISA 7.12.2, 32-bit A 16x4 (MxK): lanes 0-15 hold K=0,1 ; lanes 16-31
  // hold K=2,3 ; one row per lane.  Fragment (r,kk) covers rows 16r..16r+15,
  // cols 4kk..4kk+3:  VGPR0 = M[16r+lr][4kk+2hi], VGPR1 = M[16r+lr][4kk+2hi+1].
  v2f Af[64];
#pragma unroll
  for (int r = 0; r < 4; ++r) {
    const float* Mrow = M + (16 * r + lr) * D_DIM;
#pragma unroll
    for (int kk = 0; kk < 16; ++kk)
      Af[r * 16 + kk] = *(const v2f*)(Mrow + 4 * kk + 2 * hi);
  }

  // ---- loop over 16 query tiles (columns of T) -------------------------
#pragma unroll 1
  for (int c = 0; c < 16; ++c) {
    // B-fragments for this query tile, shared across all 4 row tiles.
    // B 4x16 (KxN): VGPR0 = B[2hi][lr] = Q[16c+lr][4kk+2hi], VGPR1 = +1.
    v2f Bf[16];
    const float* Qrow = Qs + (16 * c + lr) * D_DIM;
#pragma unroll
    for (int kk = 0; kk < 16; ++kk)
      Bf[kk] = *(const v2f*)(Qrow + 4 * kk + 2 * hi);

    float s = 0.0f;
#pragma unroll
    for (int r = 0; r < 4; ++r) {
      v8f acc = {0.f, 0.f, 0.f, 0.f, 0.f, 0.f, 0.f, 0.f};
#pragma unroll
      for (int kk = 0; kk < 16; ++kk) {
        acc = __builtin_amdgcn_wmma_f32_16x16x4_f32(
            /*neg_a=*/false, Af[r * 16 + kk],
            /*neg_b=*/false, Bf[kk],
            /*c_mod=*/(short)0, acc,
            /*reuse_a=*/false, /*reuse_b=*/false);
      }
      // D layout: VGPR i, lanes 0-15 -> (row=i, col=lane); lanes 16-31 ->
      // (row=8+i, col=lane-16).  This lane holds T[16r+i+8hi][16c+lr].
      const float* Qd = Qs + (16 * c + lr) * D_DIM + 16 * r + 8 * hi;
#pragma unroll
      for (int i = 0; i < 8; ++i) s += acc[i] * Qd[i];
    }
    // lanes l and l+16 hold the two row-halves of the same query's score
    float tot = s + __shfl_xor(s, 16, 32);
    if (hi == 0)
      S0[((size_t)b * L_DIM + leaf) * LK_DIM + 16 * c + lr] = tot;
  }
#else
  // ---- VALU fallback (keeps the file compiling if builtin is absent) ---
  for (int j = lane; j < LK_DIM; j += 32) {
    const float* qj = Qs + j * D_DIM;
    float s = 0.0f;
    for (int d = 0; d < D_DIM; ++d) {
      const float* Mr = M + d * D_DIM;
      float y = 0.0f;
      for (int e = 0; e < D_DIM; ++e) y += Mr[e] * qj[e];
      s += qj[d] * y;
    }
    S0[((size_t)b * L_DIM + leaf) * LK_DIM + j] = s;
  }
#endif
}

// ---------------------------------------------------------------------------
// Phase B: pairwise sum tree over leaves.  8 blocks (one per batch) x 256 thr.
// ---------------------------------------------------------------------------
__global__ void __launch_bounds__(256)
mt_tree_sum_kernel(float* __restrict__ ws)
{
  const int b   = blockIdx.x;
  float* S0     = ws;
  float* T      = ws + S0_FLOATS;

#pragma unroll
  for (int k = 1; k <= 9; ++k) {
    const int nk = L_DIM >> k;
    const float* src =
        (k == 1) ? (S0 + (size_t)b * L_DIM * LK_DIM)
                 : (T + lvl_off(k - 1) + (size_t)b * (L_DIM >> (k - 1)) * LK_DIM);
    float* dst = T + lvl_off(k) + (size_t)b * nk * LK_DIM;
    for (int idx = threadIdx.x; idx < nk * LK_DIM; idx += blockDim.x) {
      const int n = idx >> 8;
      const int j = idx & 255;
      dst[n * LK_DIM + j] =
          src[(2 * n) * LK_DIM + j] + src[(2 * n + 1) * LK_DIM + j];
    }
    __syncthreads();
  }
}

// ---------------------------------------------------------------------------
// Phase C: descent.  8 blocks x 256 threads = one thread per (b, j).
// ---------------------------------------------------------------------------
__global__ void __launch_bounds__(256)
mt_descend_kernel(const float* __restrict__ ws, float* __restrict__ out)
{
  const int t = blockIdx.x * blockDim.x + threadIdx.x;   // 0..2047
  const int b = t >> 8;
  const int j = t & 255;

  const float* S0 = ws;
  const float* T  = ws + S0_FLOATS;

  int c = 0;
#pragma unroll
  for (int k = 9; k >= 1; --k) {
    const int nk = L_DIM >> k;
    const float* lvl = T + lvl_off(k) + (size_t)b * nk * LK_DIM;
    const float sL = lvl[(2 * c) * LK_DIM + j];
    const float sR = lvl[(2 * c + 1) * LK_DIM + j];
    c = 2 * c + (sL < sR ? 1 : 0);
  }
  // level 0 = raw leaf scores
  const float* lvl0 = S0 + (size_t)b * L_DIM * LK_DIM;
  const float sL = lvl0[(2 * c) * LK_DIM + j];
  const float sR = lvl0[(2 * c + 1) * LK_DIM + j];
  c = 2 * c + (sL < sR ? 1 : 0);

  out[t] = (float)c;
}

// ---------------------------------------------------------------------------
extern "C" void kernel_launch(void* const* d_in, const int* in_sizes, int n_in,
                              void* d_out, int out_size, void* d_ws, size_t ws_size,
                              hipStream_t stream)
{
  (void)in_sizes; (void)n_in; (void)out_size; (void)ws_size;

  const float* leafs = (const float*)d_in[0];
  // d_in[1] = X : unused (fusion ignores x_shared)
  const float* q     = (const float*)d_in[2];
  float* out         = (float*)d_out;
  float* ws          = (float*)d_ws;

  // Phase A: 8192 leaves, 8 leaves per 256-thread block
  mt_score_kernel<<<(B_DIM * L_DIM) / 8, 256, 0, stream>>>(leafs, q, ws);
  // Phase B: sum tree
  mt_tree_sum_kernel<<<B_DIM, 256, 0, stream>>>(ws);
  // Phase C: descent
  mt_descend_kernel<<<B_DIM, 256, 0, stream>>>(ws, out);
}